// VLIDModel_54941221651317
// MI455X (gfx1250) — compile-verified
//
#include <hip/hip_runtime.h>
#include <hip/hip_bf16.h>

// ---------------------------------------------------------------------------
// Problem constants (from reference)
// ---------------------------------------------------------------------------
#define BATCH     8
#define NPROP     1000
#define NPAD      1008      // 63 * 16 row tiles
#define DIM       512
#define NCLS      1203
#define CPAD      1216      // 76 * 16 col tiles
#define ROWTILES  63
#define COLTILES  76
#define PRE_TOP   1000
#define MAX_OUT   300
#define SCORE_THR 0.01f
#define IOU_THR   0.5f
#define CLIP_R    0.65f
#define OBJ_R     0.35f
#define INV_T     100.0f    // 1 / SOFTMAX_T

typedef __attribute__((ext_vector_type(2))) float v2f;
typedef __attribute__((ext_vector_type(8))) float v8f;
typedef __attribute__((ext_vector_type(4))) int   v4i;

// ---------------------------------------------------------------------------
// CDNA5 async global->LDS copy (ASYNCcnt-tracked). Signature per compiler
// diagnostic: (v4i addrspace(1)*, v4i addrspace(3)*, imm offset, imm cpol).
// Guarded so the file compiles on toolchains lacking the builtin.
// ---------------------------------------------------------------------------
#if __has_builtin(__builtin_amdgcn_global_load_async_to_lds_b128)
#define HAVE_ASYNC_LDS 1
typedef __attribute__((address_space(1))) v4i* gv4ip;
typedef __attribute__((address_space(3))) v4i* lv4ip;
__device__ __forceinline__ void asyncCopy16(const float* src, float* ldsDst) {
    __builtin_amdgcn_global_load_async_to_lds_b128(
        (gv4ip)(v4i*)const_cast<float*>(src),
        (lv4ip)(v4i*)ldsDst, 0, 0);
}
__device__ __forceinline__ void asyncWait0() {
#if __has_builtin(__builtin_amdgcn_s_wait_asynccnt)
    __builtin_amdgcn_s_wait_asynccnt(0);
#else
    asm volatile("s_wait_asynccnt 0x0" ::: "memory");
#endif
}
#else
#define HAVE_ASYNC_LDS 0
__device__ __forceinline__ void asyncCopy16(const float* src, float* ldsDst) {
    *(float4*)ldsDst = *(const float4*)src;
}
__device__ __forceinline__ void asyncWait0() {}
#endif

// ---------------------------------------------------------------------------
// Helpers
// ---------------------------------------------------------------------------
__device__ __forceinline__ unsigned f2ord(float f) {
    unsigned u = __float_as_uint(f);
    return (u & 0x80000000u) ? ~u : (u | 0x80000000u);
}
__device__ __forceinline__ float ord2f(unsigned o) {
    unsigned u = (o & 0x80000000u) ? (o & 0x7FFFFFFFu) : ~o;
    return __uint_as_float(u);
}

__device__ __forceinline__ void waveArgMax(float& v, int& idx) {
    #pragma unroll
    for (int off = 16; off > 0; off >>= 1) {
        float ov = __shfl_xor(v, off, 32);
        int   oi = __shfl_xor(idx, off, 32);
        if (ov > v || (ov == v && oi < idx)) { v = ov; idx = oi; }
    }
}
__device__ __forceinline__ float waveSum(float s) {
    #pragma unroll
    for (int off = 16; off > 0; off >>= 1) s += __shfl_xor(s, off, 32);
    return s;
}

__device__ __forceinline__ float iou4(float4 a, float4 b) {
    float areaA = fmaxf(a.z - a.x, 0.f) * fmaxf(a.w - a.y, 0.f);
    float areaB = fmaxf(b.z - b.x, 0.f) * fmaxf(b.w - b.y, 0.f);
    float iw = fmaxf(fminf(a.z, b.z) - fmaxf(a.x, b.x), 0.f);
    float ih = fmaxf(fminf(a.w, b.w) - fmaxf(a.y, b.y), 0.f);
    float inter = iw * ih;
    float uni = areaA + areaB - inter;
    return inter / fmaxf(uni, 1e-6f);
}

// ---------------------------------------------------------------------------
// Kernel 1: L2-normalize embeddings into workspace (pen). 1 wave per row.
// pen is 16.4 MB -> L2-resident on MI455X's 192 MB L2, so the roundtrip is
// effectively free and lets kernel 2 stage A via raw async-to-LDS copies.
// ---------------------------------------------------------------------------
__global__ void normalize_kernel(const float* __restrict__ pe, float* __restrict__ pen) {
    const int wid = threadIdx.x >> 5, lane = threadIdx.x & 31;
    const int row = blockIdx.x * 8 + wid;            // 0..7999
    const float* p = pe + (size_t)row * DIM;
    float*       q = pen + (size_t)row * DIM;
    float v[DIM / 32];
    float s = 0.f;
    #pragma unroll
    for (int t = 0; t < DIM / 32; ++t) {
        v[t] = p[lane + 32 * t];
        s += v[t] * v[t];
    }
    s = waveSum(s);
    const float r = rsqrtf(s);
    #pragma unroll
    for (int t = 0; t < DIM / 32; ++t) q[lane + 32 * t] = v[t] * r;
}

// ---------------------------------------------------------------------------
// Kernel 2: fp32 WMMA GEMM  logits[b,n,c] = <pen[b,n,:], ce[c,:]>
// Block = 128 thr (4 waves). blockIdx.x -> (batch,row-tile), blockIdx.y*4+wave -> col-tile.
// A tile (16x512) staged via GLOBAL_LOAD_ASYNC_TO_LDS_B128 into a stride-516
// padded LDS tile (conflict-free ds_load_b64 frags). 128 x
// v_wmma_f32_16x16x4_f32 per output tile; full fp32 because T=0.01 amplifies
// logit error 100x inside exp. Padded cols (c>=1203) compute garbage that is
// never read downstream, so no masking VALU in the hot loop.
// ---------------------------------------------------------------------------
#define A_STRIDE 516
__global__ void wmma_gemm_kernel(const float* __restrict__ pen,
                                 const float* __restrict__ ce,
                                 float* __restrict__ logits) {
    __shared__ float lds_a[16 * A_STRIDE];           // ~33 KB
    const int tid  = threadIdx.x;
    const int lane = tid & 31;
    const int wid  = tid >> 5;
    const int bt   = blockIdx.x;                     // 0 .. BATCH*ROWTILES-1
    const int b    = bt / ROWTILES;
    const int m0   = (bt - b * ROWTILES) * 16;

    // Stage A: 2048 x 16B chunks, async DMA into padded LDS rows.
    #pragma unroll
    for (int it = 0; it < 16; ++it) {
        int idx = it * 128 + tid;
        int r   = idx >> 7;                          // tile row 0..15
        int kc  = (idx & 127) * 4;                   // K chunk start
        int n   = m0 + r;
        float* dst = &lds_a[r * A_STRIDE + kc];
        if (n < NPROP) {
            asyncCopy16(pen + ((size_t)b * NPROP + n) * DIM + kc, dst);
        } else {
            dst[0] = 0.f; dst[1] = 0.f; dst[2] = 0.f; dst[3] = 0.f;
        }
    }
    asyncWait0();
    __syncthreads();

    const int ct   = blockIdx.y * 4 + wid;           // 0..75
    const int c0   = ct * 16;
    const int cRow = c0 + (lane & 15);               // B column this lane feeds
    const float* cp = ce + (size_t)((cRow < NCLS) ? cRow : 0) * DIM;
    const int koff = (lane >> 4) * 2;                // lane half selects K pair
    const int ar   = lane & 15;                      // A row for this lane

    v8f acc = {};
    #pragma unroll 8
    for (int kb = 0; kb < DIM; kb += 4) {
        v2f a  = *(const v2f*)(&lds_a[ar * A_STRIDE + kb + koff]);   // ds_load_b64
        v2f bb = *(const v2f*)(cp + kb + koff);                      // global_load_b64 (L2-hot)
        acc = __builtin_amdgcn_wmma_f32_16x16x4_f32(
                  false, a, false, bb, (short)0, acc, false, false);
    }

    // C/D layout: VGPR v -> row m0+v (lanes 0-15) / m0+v+8 (lanes 16-31), col = c0+(lane&15)
    const int half = lane >> 4;
    const int cw   = c0 + (lane & 15);
    #pragma unroll
    for (int r = 0; r < 8; ++r) {
        int m = m0 + r + half * 8;
        logits[((size_t)b * NPAD + m) * CPAD + cw] = acc[r];
    }
}

// ---------------------------------------------------------------------------
// Kernel 3: per-row stable softmax + top-3 -> 3 candidate keys per proposal.
// One wave per proposal row. key = ordered(score)<<32 | (~flat_idx) so a
// descending uint64 sort reproduces jax top_k (value desc, index asc) order.
// ---------------------------------------------------------------------------
__global__ void score_topk_kernel(const float* __restrict__ logits,
                                  const float* __restrict__ obj,
                                  unsigned long long* __restrict__ cand) {
    const int wid  = threadIdx.x >> 5;
    const int lane = threadIdx.x & 31;
    const int row  = blockIdx.x * 8 + wid;           // 0..7999
    const int b    = row / NPROP;
    const int n    = row - b * NPROP;
    const float* base = logits + ((size_t)b * NPAD + n) * CPAD;

    float lv[38];
    #pragma unroll
    for (int t = 0; t < 38; ++t) {
        int c = lane + 32 * t;
        lv[t] = (c < NCLS) ? base[c] : -3.4e38f;
    }
    // top-1 (== row max)
    float m1 = -3.4e38f; int i1 = 0x7fffffff;
    #pragma unroll
    for (int t = 0; t < 38; ++t) { int c = lane + 32 * t; if (lv[t] > m1) { m1 = lv[t]; i1 = c; } }
    waveArgMax(m1, i1);
    // softmax denominator (max-subtracted; /T == *100)
    float s = 0.f;
    #pragma unroll
    for (int t = 0; t < 38; ++t) { int c = lane + 32 * t; if (c < NCLS) s += __expf((lv[t] - m1) * INV_T); }
    s = waveSum(s);
    // top-2
    float m2 = -3.4e38f; int i2 = 0x7fffffff;
    #pragma unroll
    for (int t = 0; t < 38; ++t) {
        int c = lane + 32 * t;
        if (c != i1 && lv[t] > m2) { m2 = lv[t]; i2 = c; }
    }
    waveArgMax(m2, i2);
    // top-3
    float m3 = -3.4e38f; int i3 = 0x7fffffff;
    #pragma unroll
    for (int t = 0; t < 38; ++t) {
        int c = lane + 32 * t;
        if (c != i1 && c != i2 && lv[t] > m3) { m3 = lv[t]; i3 = c; }
    }
    waveArgMax(m3, i3);

    const float objv = obj[row];
    if (lane < 3) {
        float mv = (lane == 0) ? m1 : ((lane == 1) ? m2 : m3);
        int   ci = (lane == 0) ? i1 : ((lane == 1) ? i2 : i3);
        float soft = __expf((mv - m1) * INV_T) / s;       // softmax value (top-3 kept by clip)
        float sc   = soft * CLIP_R + objv * OBJ_R;        // clip_k>0 always for top-3
        float enc  = (sc > SCORE_THR) ? sc : -1.0f;       // reference: where(>thr, s, -1)
        unsigned flat = (unsigned)n * (unsigned)NCLS + (unsigned)ci;
        cand[(size_t)b * (NPROP * 3) + n * 3 + lane] =
            ((unsigned long long)f2ord(enc) << 32) | (unsigned long long)(0xFFFFFFFFu - flat);
    }
}

// ---------------------------------------------------------------------------
// Kernel 4: per-image sort + serial NMS + top-300 emit. 1 block per image,
// 1024 threads, all state LDS-resident (~60 KB).
// ---------------------------------------------------------------------------
__device__ __forceinline__ void bitonicSortDesc4096(unsigned long long* keys, int tid) {
    for (unsigned kk = 2; kk <= 4096; kk <<= 1) {
        for (unsigned j = kk >> 1; j > 0; j >>= 1) {
            __syncthreads();
            for (unsigned idx = tid; idx < 4096; idx += 1024) {
                unsigned partner = idx ^ j;
                if (partner > idx) {
                    bool dirDesc = ((idx & kk) == 0);
                    unsigned long long a = keys[idx], c = keys[partner];
                    bool sw = dirDesc ? (a < c) : (a > c);
                    if (sw) { keys[idx] = c; keys[partner] = a; }
                }
            }
        }
    }
    __syncthreads();
}

__global__ void __launch_bounds__(1024)
nms_kernel(const unsigned long long* __restrict__ cand,
           const float* __restrict__ boxes,
           const int* __restrict__ image_ids,
           float* __restrict__ out) {
    __shared__ unsigned long long keys[4096];   // 32 KB
    __shared__ float4 sbox[PRE_TOP];            // 16 KB
    __shared__ float  ssc[PRE_TOP];             // 4 KB
    __shared__ int    slab[PRE_TOP];            // 4 KB
    __shared__ int    skeep[1024];              // 4 KB
    const int b = blockIdx.x;
    const int tid = threadIdx.x;

    // ---- top-1000 of the <=3000 candidates (pad to 4096 with min key) ----
    for (int i = tid; i < 4096; i += 1024)
        keys[i] = (i < NPROP * 3) ? cand[(size_t)b * (NPROP * 3) + i] : 0ull;
    bitonicSortDesc4096(keys, tid);

    if (tid < PRE_TOP) {
        unsigned long long k = keys[tid];
        float sc = ord2f((unsigned)(k >> 32));
        unsigned flat = 0xFFFFFFFFu - (unsigned)(k & 0xFFFFFFFFu);
        int pid = (int)(flat / (unsigned)NCLS);
        int lab = (int)(flat % (unsigned)NCLS);
        bool valid = sc > SCORE_THR;
        if (!valid || pid >= NPROP) pid = 0;
        const float* bp = boxes + ((size_t)b * NPROP + pid) * 4;
        sbox[tid]  = make_float4(bp[0], bp[1], bp[2], bp[3]);
        ssc[tid]   = sc;
        slab[tid]  = lab;
        skeep[tid] = valid ? 1 : 0;
    } else {
        skeep[tid] = 0;
    }

    // ---- serial greedy suppression (matches fori_loop semantics) ----
    for (int i = 0; i < PRE_TOP - 1; ++i) {
        __syncthreads();
        if (tid < PRE_TOP && tid > i) {
            if (skeep[i] && skeep[tid] && slab[tid] == slab[i]) {
                if (iou4(sbox[i], sbox[tid]) > IOU_THR) skeep[tid] = 0;
            }
        }
    }
    __syncthreads();

    // ---- top-300 of kept scores (index asc tie-break like top_k) ----
    for (int i = tid; i < 4096; i += 1024) {
        if (i < PRE_TOP) {
            float kept = skeep[i] ? ssc[i] : -1.0f;
            keys[i] = ((unsigned long long)f2ord(kept) << 32) |
                      (unsigned long long)(0xFFFFFFFFu - (unsigned)i);
        } else {
            keys[i] = 0ull;
        }
    }
    bitonicSortDesc4096(keys, tid);

    if (tid < MAX_OUT) {
        unsigned long long k = keys[tid];
        float fs = ord2f((unsigned)(k >> 32));
        unsigned fi = 0xFFFFFFFFu - (unsigned)(k & 0xFFFFFFFFu);
        bool fvalid = fs > SCORE_THR;
        int src = (fvalid && fi < PRE_TOP) ? (int)fi : 0;
        float4 fb = sbox[src];
        float* brow = out + ((size_t)b * MAX_OUT + tid) * 5;
        brow[0] = fvalid ? fb.x : 0.f;
        brow[1] = fvalid ? fb.y : 0.f;
        brow[2] = fvalid ? fb.z : 0.f;
        brow[3] = fvalid ? fb.w : 0.f;
        brow[4] = fvalid ? fs   : 0.f;
        int* labout = (int*)(out + (size_t)BATCH * MAX_OUT * 5);
        int* imgout = labout + (size_t)BATCH * MAX_OUT;
        labout[b * MAX_OUT + tid] = fvalid ? slab[src] : -1;
        imgout[b * MAX_OUT + tid] = fvalid ? image_ids[b] : -1;
    }
}

// ---------------------------------------------------------------------------
// Host launcher
// ---------------------------------------------------------------------------
extern "C" void kernel_launch(void* const* d_in, const int* in_sizes, int n_in,
                              void* d_out, int out_size, void* d_ws, size_t ws_size,
                              hipStream_t stream) {
    (void)in_sizes; (void)n_in; (void)out_size; (void)ws_size;
    const float* pe    = (const float*)d_in[0];   // (8,1000,512)
    const float* obj   = (const float*)d_in[1];   // (8,1000)
    const float* boxes = (const float*)d_in[2];   // (8,1000,4)
    const float* ce    = (const float*)d_in[3];   // (1203,512)
    const int*   img   = (const int*)d_in[4];     // (8,)
    float* out = (float*)d_out;

    char* ws = (char*)d_ws;
    const size_t penBytes    = (size_t)BATCH * NPROP * DIM * 4;      // 16,384,000
    const size_t logitsBytes = (size_t)BATCH * NPAD * CPAD * 4;      // 39,223,296
    float* pen    = (float*)ws;
    float* logits = (float*)(ws + penBytes);
    unsigned long long* cand = (unsigned long long*)(ws + penBytes + logitsBytes);

    normalize_kernel<<<NPROP, 256, 0, stream>>>(pe, pen);

    dim3 g2(BATCH * ROWTILES, COLTILES / 4);
    wmma_gemm_kernel<<<g2, 128, 0, stream>>>(pen, ce, logits);

    score_topk_kernel<<<NPROP, 256, 0, stream>>>(logits, obj, cand);

    nms_kernel<<<BATCH, 1024, 0, stream>>>(cand, boxes, img, out);
}